// MoIR_88897233092634
// MI455X (gfx1250) — compile-verified
//
#include <hip/hip_runtime.h>
#include <math.h>

// ---------------------------------------------------------------------------
// MoIR channel exchange for MI455X (gfx1250, wave32, WMMA + TDM).
//
//  K1 pool_stats : streaming pass over a,b -> pool mean, 256-token column
//                  mean, per-channel sum-of-squares.  Memory bound (~11us).
//  K2 gram_wmma  : per batch G = Xc Xc^T (256x256, K=1024) with
//                  V_WMMA_F32_16X16X4_F32, one wave per 16x16 tile.
//  K3 subspace   : TDM 2D tile load of G[:, :64] into LDS, then block
//                  subspace iteration + CholeskyQR -> orthonormal Q (256x64).
//  K4 scores     : TDM 1D load of Q into LDS; scores_d = ||Q^T Xc[:,d]||^2.
//  K5 select     : rank-count top-102 smallest scores, RMS-match scale,
//                  fold into per-(b,d) affine tables c0,c1.
//  K6 apply      : out = x*c0 + c1  (float4, dominant memory-bound ~22us).
// ---------------------------------------------------------------------------

typedef float v2f __attribute__((ext_vector_type(2)));
typedef float v8f __attribute__((ext_vector_type(8)));
typedef unsigned int u32x4 __attribute__((ext_vector_type(4)));
typedef int i32x4 __attribute__((ext_vector_type(4)));
typedef int i32x8 __attribute__((ext_vector_type(8)));

#define B_    16
#define L_    2048
#define D_    1024
#define T_    256     // TOKEN_SUB
#define Q_    64      // TOP_Q
#define K_LOW 102     // round(0.1 * 1024)
#define EPS_  1e-6f

// workspace layout (float offsets)
#define OFF_POOL   0                          // [2][B][D]
#define OFF_MU     (OFF_POOL   + 2*B_*D_)     // [2][B][D] col-mean of first 256 tokens
#define OFF_SSQ    (OFF_MU     + 2*B_*D_)     // [2][B][D] sum over L of x^2
#define OFF_SCORES (OFF_SSQ    + 2*B_*D_)     // [2][B][D]
#define OFF_C0     (OFF_SCORES + 2*B_*D_)     // [2][B][D]
#define OFF_C1     (OFF_C0     + 2*B_*D_)     // [2][B][D]
#define OFF_G      (OFF_C1     + 2*B_*D_)     // [2][B][256][256]
#define OFF_QB     (OFF_G      + 2*B_*T_*T_)  // [2][B][256][64]
// total ~11.3 MB of d_ws

// ---------------------------------------------------------------------------
// Tensor Data Mover: 2D tile (tile_w x tile_h) of 4-byte elements from a
// row-major tensor with row stride `stride_el` into contiguous LDS at lds_off.
// D# layout per CDNA5 ISA (CDNA5 ISA):
//   group0: [1:0]=count, [63:32]=lds_addr, [120:64]=global_addr, [127:126]=2
//   group1: [17:16]=data_size(2=4B), [79:48]=tensor_dim0, [111:80]=tensor_dim1,
//           [127:112]=tile_dim0, [143:128]=tile_dim1, [207:160]=dim0_stride
// Must be issued by a single wave; completion via s_wait_tensorcnt.
__device__ __forceinline__ void tdm_load_2d_f32(unsigned lds_off,
                                                const float* gsrc,
                                                unsigned tile_w, unsigned tile_h,
                                                unsigned stride_el,
                                                unsigned tensor_w, unsigned tensor_h) {
  const unsigned long long ga = (unsigned long long)(uintptr_t)gsrc;
  u32x4 g0;
  g0[0] = 1u;                                               // count=1, user mode
  g0[1] = lds_off;                                          // lds_addr (bytes)
  g0[2] = (unsigned)(ga & 0xffffffffull);                   // global_addr[31:0]
  g0[3] = (unsigned)((ga >> 32) & 0x01ffffffull) | (2u << 30); // [56:32] | type=2
  i32x8 g1;
  g1[0] = (int)(2u << 16);                                  // wg_mask=0, data_size=4B
  g1[1] = (int)((tensor_w & 0xffffu) << 16);                // tensor_dim0[15:0]
  g1[2] = (int)((tensor_w >> 16) & 0xffffu)                 // tensor_dim0[31:16]
        | (int)((tensor_h & 0xffffu) << 16);                // tensor_dim1[15:0]
  g1[3] = (int)((tensor_h >> 16) & 0xffffu)                 // tensor_dim1[31:16]
        | (int)((tile_w & 0xffffu) << 16);                  // tile_dim0
  g1[4] = (int)(tile_h & 0xffffu);                          // tile_dim1 (tile_dim2=0)
  g1[5] = (int)stride_el;                                   // tensor_dim0_stride[31:0]
  g1[6] = 0;                                                // stride hi, dim1_stride lo
  g1[7] = 0;
  const i32x4 z4 = {0, 0, 0, 0};
#if __clang_major__ >= 23
  const i32x8 z8 = {0, 0, 0, 0, 0, 0, 0, 0};
  __builtin_amdgcn_tensor_load_to_lds(g0, g1, z4, z4, z8, 0);
#else
  __builtin_amdgcn_tensor_load_to_lds(g0, g1, z4, z4, 0);
#endif
}

// ---------------------------------------------------------------------------
// K1: per (tensor,batch,channel) streaming stats.  grid=(B*4, 2), block=256.
__global__ void pool_stats_kernel(const float* __restrict__ a,
                                  const float* __restrict__ b,
                                  const unsigned char* __restrict__ amask,
                                  const unsigned char* __restrict__ bmask,
                                  float* __restrict__ ws) {
  const int z = blockIdx.y;
  const float* X = z ? b : a;
  const unsigned char* M = z ? bmask : amask;
  const int batch = blockIdx.x >> 2;
  const int d = ((blockIdx.x & 3) << 8) + threadIdx.x;
  const float* Xb = X + (size_t)batch * L_ * D_ + d;
  const unsigned char* Mb = M + batch * L_;
  float sum = 0.f, ssq = 0.f, cnt = 0.f, s256 = 0.f;
  for (int l = 0; l < L_; ++l) {
    float v = Xb[(size_t)l * D_];
    if (l + 16 < L_) __builtin_prefetch(Xb + (size_t)(l + 16) * D_, 0, 1);
    float m = Mb[l] ? 1.f : 0.f;
    sum += v * m;
    cnt += m;
    ssq += v * v;                 // reference: (dst*dst)*chan_mask sums ALL L
    if (l < T_) s256 += v;        // reference centers over first 256 tokens
  }
  const int o = (z * B_ + batch) * D_ + d;
  ws[OFF_POOL + o] = sum / fmaxf(cnt, 1.f);
  ws[OFF_MU   + o] = s256 * (1.f / (float)T_);
  ws[OFF_SSQ  + o] = ssq;
}

// ---------------------------------------------------------------------------
// K2: Gram matrix G = Xc * Xc^T per (tensor,batch) with f32 WMMA.
// One wave per 16x16 tile; K loop over D=1024 in steps of 4.
// grid = 1024 blocks * 8 waves = 2 tensors * 16 batches * 256 tiles.
__global__ void gram_wmma_kernel(const float* __restrict__ a,
                                 const float* __restrict__ b,
                                 float* __restrict__ ws) {
  const int wave = (blockIdx.x << 3) + (threadIdx.x >> 5);
  const int lane = threadIdx.x & 31;
  const int tile  = wave & 255;
  const int batch = (wave >> 8) & (B_ - 1);
  const int z     = wave >> 12;
  const int tm = tile >> 4, tn = tile & 15;
  const float* X  = (z ? b : a) + (size_t)batch * L_ * D_;
  const float* mu = ws + OFF_MU + (z * B_ + batch) * D_;
  const int lo   = lane & 15;
  const int koff = (lane >> 4) << 1;
  const float* rowA = X + (size_t)(tm * 16 + lo) * D_;
  const float* rowB = X + (size_t)(tn * 16 + lo) * D_;
  v8f c = {0.f, 0.f, 0.f, 0.f, 0.f, 0.f, 0.f, 0.f};
  for (int k = 0; k < D_; k += 4) {
    const int kk = k + koff;
    v2f A, Bv;
    A.x  = rowA[kk]     - mu[kk];
    A.y  = rowA[kk + 1] - mu[kk + 1];
    Bv.x = rowB[kk]     - mu[kk];
    Bv.y = rowB[kk + 1] - mu[kk + 1];
    c = __builtin_amdgcn_wmma_f32_16x16x4_f32(false, A, false, Bv,
                                              (short)0, c, false, false);
  }
  // C/D layout: VGPR j -> M = j (+8 for lanes 16-31), N = lane&15.
  float* G = ws + OFF_G + (size_t)(z * B_ + batch) * T_ * T_;
  const int col   = tn * 16 + lo;
  const int rbase = tm * 16 + ((lane >> 4) << 3);
#pragma unroll
  for (int j = 0; j < 8; ++j)
    G[(rbase + j) * T_ + col] = c[j];
}

// ---------------------------------------------------------------------------
// K3: subspace iteration + CholeskyQR.  1 block (256 thr) per (tensor,batch).
// dyn LDS: Q[256*64] + M/R[64*65]  (~82 KB, well under 320 KB/WGP).
// Init Q = G[:, :64] pulled in by the Tensor Data Mover (2D tile DMA).
__global__ void subspace_kernel(float* __restrict__ ws, int niter) {
  extern __shared__ float sm[];
  float* Qs = sm;                 // 256*64 : holds Q, then Y, then new Q
  float* Ms = sm + T_ * Q_;       // 64*65  : Gram of Y, then Cholesky R (upper)
  const int tb = blockIdx.x;      // z*16 + batch
  const float* G = ws + OFF_G + (size_t)tb * T_ * T_;
  const int r = threadIdx.x;      // row 0..255

  // init: Q = first 64 columns of G via TDM (tile 64 wide x 256 tall,
  // row stride 256 elements), single wave issues the descriptor.
  if (threadIdx.x < 32) {
    tdm_load_2d_f32((unsigned)(uintptr_t)(void*)Qs, G,
                    /*tile_w=*/Q_, /*tile_h=*/T_,
                    /*stride_el=*/T_, /*tensor_w=*/T_, /*tensor_h=*/T_);
    __builtin_amdgcn_s_wait_tensorcnt(0);
  }
  __syncthreads();

  float Y[Q_];
  for (int it = 0; it < niter; ++it) {
    // Y = G * Q  (each thread owns one row)
#pragma unroll
    for (int i = 0; i < Q_; ++i) Y[i] = 0.f;
    for (int t = 0; t < T_; ++t) {
      const float g = G[r * T_ + t];
#pragma unroll
      for (int i = 0; i < Q_; ++i) Y[i] += g * Qs[t * Q_ + i];
    }
    __syncthreads();
#pragma unroll
    for (int i = 0; i < Q_; ++i) Qs[r * Q_ + i] = Y[i];   // Qs := Y
    __syncthreads();

    // M = Y^T Y (upper triangle)
    for (int p = threadIdx.x; p < Q_ * Q_; p += blockDim.x) {
      const int i = p >> 6, j = p & 63;
      if (j >= i) {
        float s = 0.f;
        for (int t = 0; t < T_; ++t) s += Qs[t * Q_ + i] * Qs[t * Q_ + j];
        Ms[i * 65 + j] = s;
      }
    }
    __syncthreads();

    // Cholesky M = R^T R (serial on thread 0; 64x64 is tiny)
    if (threadIdx.x == 0) {
      for (int j = 0; j < Q_; ++j) {
        float s = Ms[j * 65 + j];
        for (int t = 0; t < j; ++t) { const float v = Ms[t * 65 + j]; s -= v * v; }
        s = sqrtf(fmaxf(s, 1e-12f));
        Ms[j * 65 + j] = s;
        const float inv = 1.f / s;
        for (int cc = j + 1; cc < Q_; ++cc) {
          float v = Ms[j * 65 + cc];
          for (int t = 0; t < j; ++t) v -= Ms[t * 65 + j] * Ms[t * 65 + cc];
          Ms[j * 65 + cc] = v * inv;
        }
      }
    }
    __syncthreads();

    // Q = Y R^{-1}: per-row forward substitution  x_j = (y_j - sum x_i R_ij)/R_jj
#pragma unroll
    for (int i = 0; i < Q_; ++i) Y[i] = Qs[r * Q_ + i];
    float x[Q_];
#pragma unroll
    for (int j = 0; j < Q_; ++j) {
      float v = Y[j];
#pragma unroll
      for (int i = 0; i < j; ++i) v -= x[i] * Ms[i * 65 + j];
      x[j] = v / Ms[j * 65 + j];
    }
    __syncthreads();
#pragma unroll
    for (int j = 0; j < Q_; ++j) Qs[r * Q_ + j] = x[j];
    __syncthreads();
  }

  float* Qo = ws + OFF_QB + (size_t)tb * T_ * Q_;
#pragma unroll
  for (int i = 0; i < Q_; ++i) Qo[r * Q_ + i] = Qs[r * Q_ + i];
}

// ---------------------------------------------------------------------------
// K4: scores_d = || Q^T Xc[:,d] ||^2.  grid = 2*16*4 blocks, 256 thr, one d each.
// Q (contiguous 64 KB) staged into LDS by a 1D TDM transfer.
__global__ void scores_kernel(const float* __restrict__ a,
                              const float* __restrict__ b,
                              float* __restrict__ ws) {
  extern __shared__ float Qs[];   // 256*64
  const int blk   = blockIdx.x;
  const int z     = blk >> 6;
  const int batch = (blk >> 2) & 15;
  const int chunk = blk & 3;
  const float* Qg = ws + OFF_QB + (size_t)(z * B_ + batch) * T_ * Q_;
  if (threadIdx.x < 32) {
    tdm_load_2d_f32((unsigned)(uintptr_t)(void*)Qs, Qg,
                    /*tile_w=*/T_ * Q_, /*tile_h=*/1,
                    /*stride_el=*/T_ * Q_, /*tensor_w=*/T_ * Q_, /*tensor_h=*/1);
    __builtin_amdgcn_s_wait_tensorcnt(0);
  }
  __syncthreads();

  const int d = (chunk << 8) + threadIdx.x;
  const float* X = (z ? b : a) + (size_t)batch * L_ * D_ + d;
  const float mu = ws[OFF_MU + (z * B_ + batch) * D_ + d];
  float p[Q_];
#pragma unroll
  for (int i = 0; i < Q_; ++i) p[i] = 0.f;
  for (int t = 0; t < T_; ++t) {
    const float xv = X[(size_t)t * D_] - mu;
#pragma unroll
    for (int i = 0; i < Q_; ++i) p[i] += xv * Qs[t * Q_ + i];
  }
  float s = 0.f;
#pragma unroll
  for (int i = 0; i < Q_; ++i) s += p[i] * p[i];
  ws[OFF_SCORES + (z * B_ + batch) * D_ + d] = s;
}

// ---------------------------------------------------------------------------
// K5: rank-count selection of the 102 smallest scores (jax top_k tie order:
// earlier index wins), RMS-matched scale, fold into affine tables:
//   out = x*(1 - m*alpha) + m*alpha*scale*src_pool  ->  c0, c1.
__global__ void select_tables_kernel(const float* __restrict__ alpha_a,
                                     const float* __restrict__ alpha_b,
                                     float* __restrict__ ws) {
  __shared__ float s[D_];
  __shared__ float accD, accS, scaleSh;
  const int z     = blockIdx.x >> 4;    // 0: a_out (dst=a, src=b_pool)
  const int batch = blockIdx.x & 15;
  const float* sc    = ws + OFF_SCORES + (z * B_ + batch) * D_;
  const float* ssqD  = ws + OFF_SSQ    + (z * B_ + batch) * D_;
  const float* poolS = ws + OFF_POOL   + ((1 - z) * B_ + batch) * D_;
  const float* al = z ? alpha_b : alpha_a;

  for (int d = threadIdx.x; d < D_; d += blockDim.x) s[d] = sc[d];
  if (threadIdx.x == 0) { accD = 0.f; accS = 0.f; }
  __syncthreads();

  float mloc[4];
#pragma unroll
  for (int q = 0; q < 4; ++q) {
    const int d = threadIdx.x + (q << 8);
    const float sd = s[d];
    int rank = 0;
    for (int e = 0; e < D_; ++e) {
      const float se = s[e];
      rank += (se < sd) || (se == sd && e < d);
    }
    const float m = (rank < K_LOW) ? 1.f : 0.f;
    mloc[q] = m;
    if (m != 0.f) {
      atomicAdd(&accD, ssqD[d]);            // LDS ds_add_f32
      const float pv = poolS[d];
      atomicAdd(&accS, pv * pv);
    }
  }
  __syncthreads();
  if (threadIdx.x == 0) {
    const float rmsD = sqrtf(accD / (float)(L_ * K_LOW) + EPS_);
    const float rmsS = sqrtf(accS / (float)K_LOW + EPS_);
    scaleSh = fminf(fmaxf(rmsD / (rmsS + EPS_), 0.1f), 10.f);
  }
  __syncthreads();
  const float scl = scaleSh;
#pragma unroll
  for (int q = 0; q < 4; ++q) {
    const int d = threadIdx.x + (q << 8);
    const float alpha = 1.f / (1.f + expf(-al[d]));
    const float ma = mloc[q] * alpha;
    const int o = (z * B_ + batch) * D_ + d;
    ws[OFF_C0 + o] = 1.f - ma;
    ws[OFF_C1 + o] = ma * scl * poolS[d];
  }
}

// ---------------------------------------------------------------------------
// K6: out = x*c0[b,d] + c1[b,d], float4.  grid=(32768, 2), block=256.
__global__ void apply_kernel(const float* __restrict__ a,
                             const float* __restrict__ b,
                             const float* __restrict__ ws,
                             float* __restrict__ out) {
  const int z = blockIdx.y;
  const size_t idx4 = ((size_t)blockIdx.x * blockDim.x + threadIdx.x) << 2;
  const float* X = z ? b : a;
  const int batch = (int)(idx4 >> 21);        // L_*D_ = 2^21
  const int d     = (int)(idx4 & (D_ - 1));   // 4-aligned within a row
  const float4 xv = *(const float4*)(X + idx4);
  const float4 c0 = *(const float4*)(ws + OFF_C0 + (z * B_ + batch) * D_ + d);
  const float4 c1 = *(const float4*)(ws + OFF_C1 + (z * B_ + batch) * D_ + d);
  float4 o;
  o.x = xv.x * c0.x + c1.x;
  o.y = xv.y * c0.y + c1.y;
  o.z = xv.z * c0.z + c1.z;
  o.w = xv.w * c0.w + c1.w;
  *(float4*)(out + (size_t)z * B_ * L_ * D_ + idx4) = o;
}

// ---------------------------------------------------------------------------
extern "C" void kernel_launch(void* const* d_in, const int* in_sizes, int n_in,
                              void* d_out, int out_size, void* d_ws, size_t ws_size,
                              hipStream_t stream) {
  (void)in_sizes; (void)n_in; (void)out_size; (void)ws_size;
  const float* a = (const float*)d_in[0];
  const float* b = (const float*)d_in[1];
  const unsigned char* am = (const unsigned char*)d_in[2];
  const unsigned char* bm = (const unsigned char*)d_in[3];
  const float* ala = (const float*)d_in[4];
  const float* alb = (const float*)d_in[5];
  float* ws  = (float*)d_ws;
  float* out = (float*)d_out;

  pool_stats_kernel<<<dim3(B_ * 4, 2), 256, 0, stream>>>(a, b, am, bm, ws);
  gram_wmma_kernel<<<dim3(1024), 256, 0, stream>>>(a, b, ws);
  subspace_kernel<<<dim3(32), 256, (T_ * Q_ + Q_ * 65) * sizeof(float), stream>>>(ws, 6);
  scores_kernel<<<dim3(128), 256, T_ * Q_ * sizeof(float), stream>>>(a, b, ws);
  select_tables_kernel<<<dim3(32), 256, 0, stream>>>(ala, alb, ws);
  apply_kernel<<<dim3((B_ * L_ * D_) / 4 / 256, 2), 256, 0, stream>>>(a, b, ws, out);
}